// GNNPolicyFraudDetector_3418793967835
// MI455X (gfx1250) — compile-verified
//
#include <hip/hip_runtime.h>
#include <hip/hip_bf16.h>
#include <math.h>

// ---------------------------------------------------------------------------
// GCN 3-layer forward for MI455X (gfx1250, wave32).
// Dense GEMMs use V_WMMA_F32_16X16X4_F32 (full fp32 semantics).
// Edge scatter is the roofline-dominant stage (~3 GB, L2-resident on a
// 192MB-L2 part): float4-vectorized gather + f32 global atomics, with
// per-edge norm precomputed once.
// ---------------------------------------------------------------------------

typedef float v2f __attribute__((ext_vector_type(2)));
typedef float v8f __attribute__((ext_vector_type(8)));

// ---------------- normalization kernels ------------------------------------

__global__ void k_fill_ones(float* __restrict__ p, int n) {
    int i = blockIdx.x * blockDim.x + threadIdx.x;
    if (i < n) p[i] = 1.0f;   // self-loop contributes 1 to every degree
}

__global__ void k_deg_accum(const int* __restrict__ dst, float* __restrict__ deg, int E) {
    int e = blockIdx.x * blockDim.x + threadIdx.x;
    if (e < E) atomicAdd(&deg[dst[e]], 1.0f);
}

__global__ void k_rsqrt_inplace(float* __restrict__ p, int n) {
    int i = blockIdx.x * blockDim.x + threadIdx.x;
    if (i < n) p[i] = rsqrtf(p[i]);   // deg >= 1 always (self loops)
}

// norm[e] = dinv[src[e]] * dinv[dst[e]]  (computed once, reused by all layers)
__global__ void k_edge_norm(const int* __restrict__ src, const int* __restrict__ dst,
                            const float* __restrict__ dinv, float* __restrict__ norm, int E) {
    int e = blockIdx.x * blockDim.x + threadIdx.x;
    if (e < E) norm[e] = dinv[src[e]] * dinv[dst[e]];
}

// ---------------- aggregation kernels --------------------------------------

// agg[i][f] = h[i][f] * dinv[i]^2  (self-loop message; also initializes agg)
// float4 version: one thread per 4 features. shift4 = log2(F/4).
__global__ void k_init_agg4(const float4* __restrict__ h, const float* __restrict__ dinv,
                            float4* __restrict__ agg, int total4, int shift4) {
    int i = blockIdx.x * blockDim.x + threadIdx.x;
    if (i >= total4) return;
    int node = i >> shift4;
    float w = dinv[node];
    w *= w;
    float4 v = h[i];
    v.x *= w; v.y *= w; v.z *= w; v.w *= w;
    agg[i] = v;
}

// scalar version for the 2-wide final layer
__global__ void k_init_agg(const float* __restrict__ h, const float* __restrict__ dinv,
                           float* __restrict__ agg, int total, int shift) {
    int i = blockIdx.x * blockDim.x + threadIdx.x;
    if (i >= total) return;
    int node = i >> shift;
    float w = dinv[node];
    agg[i] = h[i] * w * w;
}

// Vectorized edge scatter: each thread handles 4 consecutive features of one
// edge.  F = 1<<LOGF features; threads-per-edge = F/4.
template <int LOGF>
__global__ void k_edge_scatter4(const int* __restrict__ src, const int* __restrict__ dst,
                                const float* __restrict__ norm, const float* __restrict__ h,
                                float* __restrict__ agg, long long total) {
    long long t = (long long)blockIdx.x * blockDim.x + threadIdx.x;
    if (t >= total) return;
    constexpr int TPE_LOG = LOGF - 2;                 // log2(threads per edge)
    long long e = t >> TPE_LOG;
    int f = (int)(t & (long long)((1 << TPE_LOG) - 1)) << 2;
    int s = src[e];
    int d = dst[e];
    float w = norm[e];
    float4 v = *(const float4*)(h + ((size_t)s << LOGF) + f);
    float* ap = agg + ((size_t)d << LOGF) + f;
    atomicAdd(ap + 0, v.x * w);
    atomicAdd(ap + 1, v.y * w);
    atomicAdd(ap + 2, v.z * w);
    atomicAdd(ap + 3, v.w * w);
}

// F=2 scatter: one thread per edge, float2 payload
__global__ void k_edge_scatter_f2(const int* __restrict__ src, const int* __restrict__ dst,
                                  const float* __restrict__ norm, const float* __restrict__ h,
                                  float* __restrict__ agg, int E) {
    int e = blockIdx.x * blockDim.x + threadIdx.x;
    if (e >= E) return;
    int s = src[e];
    int d = dst[e];
    float w = norm[e];
    float2 v = *(const float2*)(h + (size_t)s * 2);
    atomicAdd(&agg[(size_t)d * 2 + 0], v.x * w);
    atomicAdd(&agg[(size_t)d * 2 + 1], v.y * w);
}

// a[i] = relu(a[i] + b[f]) — float4, fmask4 = (F/4 - 1)
__global__ void k_bias_relu4(float4* __restrict__ a, const float4* __restrict__ b,
                             int total4, int fmask4) {
    int i = blockIdx.x * blockDim.x + threadIdx.x;
    if (i >= total4) return;
    float4 v = a[i];
    float4 bb = b[i & fmask4];
    v.x = fmaxf(v.x + bb.x, 0.0f);
    v.y = fmaxf(v.y + bb.y, 0.0f);
    v.z = fmaxf(v.z + bb.z, 0.0f);
    v.w = fmaxf(v.w + bb.w, 0.0f);
    a[i] = v;
}

// ---------------- WMMA fp32 GEMM: C[M x N] = A[M x K] * W[K x N] -----------
// One wave per 16-row tile of M. N multiple of 16 (<=64), K multiple of 4.
// VGPR layouts per CDNA5 ISA 7.12.2:
//   A 16x4:  lane<16 -> K={k0,k0+1}, lane>=16 -> K={k0+2,k0+3}; vgpr = K within half
//   B 4x16:  mirrored (col = lane&15)
//   C 16x16: vgpr r -> row r (lanes 0-15) / row r+8 (lanes 16-31), col = lane&15
template <int K, int N>
__global__ void k_wmma_gemm(const float* __restrict__ A, const float* __restrict__ W,
                            float* __restrict__ C, int M) {
    constexpr int NT = N / 16;
    int wave = (int)((blockIdx.x * blockDim.x + threadIdx.x) >> 5);
    int lane = threadIdx.x & 31;
    int tiles = M >> 4;                 // M is a multiple of 16 (100000 = 6250*16)
    if (wave >= tiles) return;          // wave-uniform: EXEC stays all-ones
    int half = lane >> 4;               // 0 or 1
    int m    = lane & 15;

    const float* arow = A + (size_t)(wave * 16 + m) * K;

    v8f acc[NT] = {};                   // zero accumulators

    for (int k0 = 0; k0 < K; k0 += 4) {
        int ka = k0 + 2 * half;
        v2f a;
        a.x = arow[ka];
        a.y = arow[ka + 1];
#pragma unroll
        for (int nt = 0; nt < NT; ++nt) {
            v2f b;
            b.x = W[(size_t)ka * N + nt * 16 + m];
            b.y = W[(size_t)(ka + 1) * N + nt * 16 + m];
            acc[nt] = __builtin_amdgcn_wmma_f32_16x16x4_f32(
                /*neg_a=*/false, a, /*neg_b=*/false, b,
                /*c_mod=*/(short)0, acc[nt], /*reuse_a=*/false, /*reuse_b=*/false);
        }
    }

#pragma unroll
    for (int nt = 0; nt < NT; ++nt) {
#pragma unroll
        for (int r = 0; r < 8; ++r) {
            int row = wave * 16 + r + 8 * half;
            C[(size_t)row * N + nt * 16 + m] = acc[nt][r];
        }
    }
}

// ---------------- tiny layer-3 GEMM (K=32, N=2): scalar per node -----------
__global__ void k_gemm3(const float* __restrict__ act, const float* __restrict__ W3,
                        float* __restrict__ h3, int n) {
    int i = blockIdx.x * blockDim.x + threadIdx.x;
    if (i >= n) return;
    float z0 = 0.0f, z1 = 0.0f;
    const float* a = act + (size_t)i * 32;
#pragma unroll
    for (int k = 0; k < 32; ++k) {
        float v = a[k];
        z0 = fmaf(v, W3[k * 2 + 0], z0);
        z1 = fmaf(v, W3[k * 2 + 1], z1);
    }
    h3[(size_t)i * 2 + 0] = z0;
    h3[(size_t)i * 2 + 1] = z1;
}

// ---------------- final bias + 2-class log-softmax -------------------------
__global__ void k_logsoftmax2(const float* __restrict__ agg3, const float* __restrict__ b3,
                              float* __restrict__ out, int n) {
    int i = blockIdx.x * blockDim.x + threadIdx.x;
    if (i >= n) return;
    float z0 = agg3[(size_t)i * 2 + 0] + b3[0];
    float z1 = agg3[(size_t)i * 2 + 1] + b3[1];
    float mx = fmaxf(z0, z1);
    float lse = mx + logf(expf(z0 - mx) + expf(z1 - mx));
    out[(size_t)i * 2 + 0] = z0 - lse;
    out[(size_t)i * 2 + 1] = z1 - lse;
}

// ---------------------------------------------------------------------------

static inline int cdiv(long long a, int b) { return (int)((a + b - 1) / b); }

extern "C" void kernel_launch(void* const* d_in, const int* in_sizes, int n_in,
                              void* d_out, int out_size, void* d_ws, size_t ws_size,
                              hipStream_t stream) {
    const float* x  = (const float*)d_in[0];
    const int*   ei = (const int*)d_in[1];     // edge_index [2, E] (int32 on device)
    const float* W1 = (const float*)d_in[2];
    const float* b1 = (const float*)d_in[3];
    const float* W2 = (const float*)d_in[4];
    const float* b2 = (const float*)d_in[5];
    const float* W3 = (const float*)d_in[6];
    const float* b3 = (const float*)d_in[7];

    const int NFEAT = 128;
    const int Nn = in_sizes[0] / NFEAT;        // 100000
    const int E  = in_sizes[1] / 2;            // 3200000
    const int* src = ei;                       // edge_index[0]
    const int* dst = ei + E;                   // edge_index[1]

    // workspace layout (floats); buffers reused across layers
    float* dinv = (float*)d_ws;                          // Nn (deg, then rsqrt in place)
    float* bufA = dinv + 102400;                         // Nn*64: h1 -> h2 -> h3
    float* bufB = bufA + (size_t)Nn * 64;                // Nn*64: agg1/act1 -> agg2/act2 -> agg3
    float* norm = bufB + (size_t)Nn * 64;                // E edge norms

    const int T = 256;

    // ---- symmetric normalization: dinv = rsqrt(1 + indegree), edge norms ----
    k_fill_ones<<<cdiv(Nn, T), T, 0, stream>>>(dinv, Nn);
    k_deg_accum<<<cdiv(E, T), T, 0, stream>>>(dst, dinv, E);
    k_rsqrt_inplace<<<cdiv(Nn, T), T, 0, stream>>>(dinv, Nn);
    k_edge_norm<<<cdiv(E, T), T, 0, stream>>>(src, dst, dinv, norm, E);

    const int gemm_blocks = cdiv((long long)(Nn / 16) * 32, T);

    // ---- layer 1: h1 = x @ W1 (128 -> 64), aggregate, +b1, relu ----
    k_wmma_gemm<128, 64><<<gemm_blocks, T, 0, stream>>>(x, W1, bufA, Nn);
    k_init_agg4<<<cdiv((long long)Nn * 16, T), T, 0, stream>>>(
        (const float4*)bufA, dinv, (float4*)bufB, Nn * 16, 4);
    k_edge_scatter4<6><<<cdiv((long long)E * 16, T), T, 0, stream>>>(
        src, dst, norm, bufA, bufB, (long long)E * 16);
    k_bias_relu4<<<cdiv((long long)Nn * 16, T), T, 0, stream>>>(
        (float4*)bufB, (const float4*)b1, Nn * 16, 15);

    // ---- layer 2: h2 = act1 @ W2 (64 -> 32), aggregate, +b2, relu ----
    k_wmma_gemm<64, 32><<<gemm_blocks, T, 0, stream>>>(bufB, W2, bufA, Nn);
    k_init_agg4<<<cdiv((long long)Nn * 8, T), T, 0, stream>>>(
        (const float4*)bufA, dinv, (float4*)bufB, Nn * 8, 3);
    k_edge_scatter4<5><<<cdiv((long long)E * 8, T), T, 0, stream>>>(
        src, dst, norm, bufA, bufB, (long long)E * 8);
    k_bias_relu4<<<cdiv((long long)Nn * 8, T), T, 0, stream>>>(
        (float4*)bufB, (const float4*)b2, Nn * 8, 7);

    // ---- layer 3: h3 = act2 @ W3 (32 -> 2), aggregate, +b3, log_softmax ----
    k_gemm3<<<cdiv(Nn, T), T, 0, stream>>>(bufB, W3, bufA, Nn);
    k_init_agg<<<cdiv((long long)Nn * 2, T), T, 0, stream>>>(bufA, dinv, bufB, Nn * 2, 1);
    k_edge_scatter_f2<<<cdiv(E, T), T, 0, stream>>>(src, dst, norm, bufA, bufB, E);
    k_logsoftmax2<<<cdiv(Nn, T), T, 0, stream>>>(bufB, b3, (float*)d_out, Nn);
}